// Mgcnlinear_32822140076323
// MI455X (gfx1250) — compile-verified
//
#include <hip/hip_runtime.h>

#define NN 8192
#define CC 256
#define KNBR 3
#define R2V 9.0f

typedef float v2f __attribute__((ext_vector_type(2)));
typedef float v8f __attribute__((ext_vector_type(8)));

__device__ __forceinline__ v8f wmma4(v2f a, v2f b, v8f c) {
    // D = A(16x4,f32) * B(4x16,f32) + C(16x16,f32), exact f32 path
    return __builtin_amdgcn_wmma_f32_16x16x4_f32(false, a, false, b, (short)0, c,
                                                 false, false);
}

// ---------------- kernel 1: rowwise softmax + squared norm ----------------
__global__ __launch_bounds__(256) void k_softmax(const float* __restrict__ x,
                                                 float* __restrict__ p,
                                                 float* __restrict__ sq) {
    __shared__ float red[256];
    const int row = blockIdx.x;
    const int t = threadIdx.x;
    float v = x[row * CC + t];
    red[t] = v; __syncthreads();
    for (int s = 128; s > 0; s >>= 1) { if (t < s) red[t] = fmaxf(red[t], red[t + s]); __syncthreads(); }
    float m = red[0]; __syncthreads();
    float e = expf(v - m);
    red[t] = e; __syncthreads();
    for (int s = 128; s > 0; s >>= 1) { if (t < s) red[t] += red[t + s]; __syncthreads(); }
    float sum = red[0]; __syncthreads();
    float pv = e / sum;
    p[row * CC + t] = pv;
    red[t] = pv * pv; __syncthreads();
    for (int s = 128; s > 0; s >>= 1) { if (t < s) red[t] += red[t + s]; __syncthreads(); }
    if (t == 0) sq[row] = red[0];
}

// ---------------- kernel 2: Y = relu(X @ W^T + bias), WMMA f32 ----------------
__global__ __launch_bounds__(256) void k_gemm_relu(const float* __restrict__ X,
                                                   const float* __restrict__ W,
                                                   const float* __restrict__ bias,
                                                   float* __restrict__ Y) {
    __shared__ float Xs[16 * CC];
    const int m0 = blockIdx.x * 16;
    const int t = threadIdx.x;
    for (int i = t; i < 16 * CC; i += 256) Xs[i] = X[(size_t)m0 * CC + i];
    __syncthreads();
    const int lane = t & 31, w = t >> 5;
    const int hi = lane >> 4, ln = lane & 15;
    const float* ar = Xs + ln * CC + 2 * hi;
    for (int nt = w; nt < CC / 16; nt += 8) {
        const int n = nt * 16 + ln;
        const float* wr = W + (size_t)n * CC + 2 * hi;
        v8f acc = {};
        for (int k0 = 0; k0 < CC; k0 += 4) {
            v2f a = *(const v2f*)(ar + k0);
            v2f b = *(const v2f*)(wr + k0);
            acc = wmma4(a, b, acc);
        }
        const float bv = bias[n];
#pragma unroll
        for (int v = 0; v < 8; ++v) {
            float r = acc[v] + bv;
            Y[(size_t)(m0 + v + 8 * hi) * CC + n] = fmaxf(r, 0.0f);
        }
    }
}

// ------- kernel 3: fused pairwise-distance GEMM + per-row top-3 (radius knn) -------
__global__ __launch_bounds__(256) void k_topk(const float* __restrict__ p,
                                              const float* __restrict__ sq,
                                              int* __restrict__ nidx,
                                              float* __restrict__ nmask) {
    __shared__ float Ps[16 * CC];
    __shared__ float sqM[16];
    __shared__ float Dt[8 * 256];      // per-wave 16x16 D tile
    __shared__ float mv[8][16][3];
    __shared__ int   mi[8][16][3];
    const int m0 = blockIdx.x * 16;
    const int t = threadIdx.x;
    for (int i = t; i < 16 * CC; i += 256) Ps[i] = p[(size_t)m0 * CC + i];
    if (t < 16) sqM[t] = sq[m0 + t];
    __syncthreads();
    const int lane = t & 31, w = t >> 5;
    const int hi = lane >> 4, ln = lane & 15;
    float sqm[8];
#pragma unroll
    for (int v = 0; v < 8; ++v) sqm[v] = sqM[v + 8 * hi];
    const float* ar = Ps + ln * CC + 2 * hi;
    float* Dw = Dt + w * 256;
    float b0 = 1e30f, b1v = 1e30f, b2v = 1e30f;
    int i0 = 0, i1 = 0, i2 = 0;
    for (int jt = w; jt < NN / 16; jt += 8) {
        const int colbase = jt * 16;
        const float* br = p + (size_t)(colbase + ln) * CC + 2 * hi;
        const float sqn = sq[colbase + ln];
        v8f acc = {};
        for (int k0 = 0; k0 < CC; k0 += 4) {
            v2f a = *(const v2f*)(ar + k0);
            v2f b = *(const v2f*)(br + k0);
            acc = wmma4(a, b, acc);
        }
#pragma unroll
        for (int v = 0; v < 8; ++v) {
            float d = sqm[v] + sqn - 2.0f * acc[v];
            Dw[(v + 8 * hi) * 16 + ln] = d;
        }
        __syncthreads();
        if (lane < 16) {
            for (int c = 0; c < 16; ++c) {
                float d = Dw[lane * 16 + c];
                int j = colbase + c;  // candidates visited in increasing j per wave
                if (d < b0)       { b2v = b1v; i2 = i1; b1v = b0; i1 = i0; b0 = d; i0 = j; }
                else if (d < b1v) { b2v = b1v; i2 = i1; b1v = d; i1 = j; }
                else if (d < b2v) { b2v = d; i2 = j; }
            }
        }
        __syncthreads();
    }
    if (lane < 16) {
        mv[w][lane][0] = b0;  mv[w][lane][1] = b1v; mv[w][lane][2] = b2v;
        mi[w][lane][0] = i0;  mi[w][lane][1] = i1;  mi[w][lane][2] = i2;
    }
    __syncthreads();
    if (w == 0 && lane < 16) {
        const int row = lane;
        float cv[24]; int ci[24];
#pragma unroll
        for (int ww = 0; ww < 8; ++ww)
            for (int s = 0; s < 3; ++s) { cv[ww * 3 + s] = mv[ww][row][s]; ci[ww * 3 + s] = mi[ww][row][s]; }
        for (int s = 0; s < KNBR; ++s) {
            float bv = cv[0]; int bj = ci[0]; int bk = 0;
            for (int q = 1; q < 24; ++q) {
                float vq = cv[q]; int jq = ci[q];
                if (vq < bv || (vq == bv && jq < bj)) { bv = vq; bj = jq; bk = q; }
            }
            cv[bk] = 1e30f;
            nidx[(m0 + row) * KNBR + s] = bj;
            nmask[(m0 + row) * KNBR + s] = (bv <= R2V) ? 1.0f : 0.0f;
        }
    }
}

// ---------------- kernel 4: neighbor gather-sum ----------------
__global__ __launch_bounds__(256) void k_agg(const float* __restrict__ h,
                                             const int* __restrict__ nidx,
                                             const float* __restrict__ nmask,
                                             float* __restrict__ agg) {
    const int row = blockIdx.x;
    const int t = threadIdx.x;
    const int j0 = nidx[row * 3 + 0], j1 = nidx[row * 3 + 1], j2 = nidx[row * 3 + 2];
    const float m0v = nmask[row * 3 + 0], m1v = nmask[row * 3 + 1], m2v = nmask[row * 3 + 2];
    agg[(size_t)row * CC + t] = h[(size_t)j0 * CC + t] * m0v +
                                h[(size_t)j1 * CC + t] * m1v +
                                h[(size_t)j2 * CC + t] * m2v;
}

// ------- kernel 5: x1 = relu(agg@Wrel^T + brel + h@Wroot^T), dual-WMMA fused -------
__global__ __launch_bounds__(256) void k_gemm2_relu(const float* __restrict__ A1,
                                                    const float* __restrict__ Wrel,
                                                    const float* __restrict__ brel,
                                                    const float* __restrict__ A2,
                                                    const float* __restrict__ Wroot,
                                                    float* __restrict__ Y) {
    __shared__ float As[16 * CC];
    __shared__ float Hs[16 * CC];
    const int m0 = blockIdx.x * 16;
    const int t = threadIdx.x;
    for (int i = t; i < 16 * CC; i += 256) {
        As[i] = A1[(size_t)m0 * CC + i];
        Hs[i] = A2[(size_t)m0 * CC + i];
    }
    __syncthreads();
    const int lane = t & 31, w = t >> 5;
    const int hi = lane >> 4, ln = lane & 15;
    const float* a1r = As + ln * CC + 2 * hi;
    const float* a2r = Hs + ln * CC + 2 * hi;
    for (int nt = w; nt < CC / 16; nt += 8) {
        const int n = nt * 16 + ln;
        const float* w1r = Wrel + (size_t)n * CC + 2 * hi;
        const float* w2r = Wroot + (size_t)n * CC + 2 * hi;
        v8f acc = {};
        for (int k0 = 0; k0 < CC; k0 += 4) {
            v2f a1 = *(const v2f*)(a1r + k0);
            v2f b1 = *(const v2f*)(w1r + k0);
            acc = wmma4(a1, b1, acc);
            v2f a2 = *(const v2f*)(a2r + k0);
            v2f b2 = *(const v2f*)(w2r + k0);
            acc = wmma4(a2, b2, acc);
        }
        const float bv = brel[n];
#pragma unroll
        for (int v = 0; v < 8; ++v) {
            float r = acc[v] + bv;
            Y[(size_t)(m0 + v + 8 * hi) * CC + n] = fmaxf(r, 0.0f);
        }
    }
}

// ---------------- kernel 6: out = x1 @ W2^T + b2 (wave-per-row) ----------------
__global__ __launch_bounds__(256) void k_out(const float* __restrict__ x1,
                                             const float* __restrict__ W2,
                                             const float* __restrict__ b2,
                                             float* __restrict__ out) {
    const int w = threadIdx.x >> 5, lane = threadIdx.x & 31;
    const int row = blockIdx.x * 8 + w;
    float a0 = 0.0f, a1 = 0.0f;
    for (int k = lane; k < CC; k += 32) {
        float xv = x1[(size_t)row * CC + k];
        a0 += xv * W2[k];
        a1 += xv * W2[CC + k];
    }
    for (int m = 16; m > 0; m >>= 1) {
        a0 += __shfl_xor(a0, m, 32);
        a1 += __shfl_xor(a1, m, 32);
    }
    if (lane == 0) {
        out[row * 2 + 0] = a0 + b2[0];
        out[row * 2 + 1] = a1 + b2[1];
    }
}

extern "C" void kernel_launch(void* const* d_in, const int* in_sizes, int n_in,
                              void* d_out, int out_size, void* d_ws, size_t ws_size,
                              hipStream_t stream) {
    const float* x     = (const float*)d_in[0];
    const float* W1    = (const float*)d_in[1];
    const float* b1    = (const float*)d_in[2];
    const float* Wrel  = (const float*)d_in[3];
    const float* brel  = (const float*)d_in[4];
    const float* Wroot = (const float*)d_in[5];
    const float* W2    = (const float*)d_in[6];
    const float* b2    = (const float*)d_in[7];

    float* out = (float*)d_out;      // [8192*2]
    float* x1  = out + (size_t)NN * 2;  // [8192*256]

    char* ws = (char*)d_ws;
    float* p     = (float*)ws;                                   // 8 MB (aliased by agg later)
    float* h     = (float*)(ws + (size_t)NN * CC * 4);           // 8 MB
    float* sq    = (float*)(ws + (size_t)2 * NN * CC * 4);       // 32 KB
    int*   nidx  = (int*)  (ws + (size_t)2 * NN * CC * 4 + (size_t)NN * 4);
    float* nmask = (float*)(ws + (size_t)2 * NN * CC * 4 + (size_t)NN * 4 + (size_t)NN * KNBR * 4);
    float* agg   = p;  // p is dead after k_topk; reuse its space

    k_softmax  <<<NN,       256, 0, stream>>>(x, p, sq);
    k_gemm_relu<<<NN / 16,  256, 0, stream>>>(x, W1, b1, h);
    k_topk     <<<NN / 16,  256, 0, stream>>>(p, sq, nidx, nmask);
    k_agg      <<<NN,       256, 0, stream>>>(h, nidx, nmask, agg);
    k_gemm2_relu<<<NN / 16, 256, 0, stream>>>(agg, Wrel, brel, h, Wroot, x1);
    k_out      <<<NN / 8,   256, 0, stream>>>(x1, W2, b2, out);
}